// SSIM_17927193493530
// MI455X (gfx1250) — compile-verified
//
#include <hip/hip_runtime.h>

// SSIM (11x11 Gaussian, depthwise) on MI455X / gfx1250.
// Separable conv done as banded-matrix multiplies on V_WMMA_F32_16X16X4_F32.

typedef __attribute__((ext_vector_type(2))) float v2f;
typedef __attribute__((ext_vector_type(8))) float v8f;

#define IMG_H 512
#define IMG_W 512
#define NC    96          // N*C = 32*3 planes
#define TPB   8           // tiles processed per block
#define KCH   7           // K chunks: 28 / 4

// select w[d] for d in [0,10], else 0 — fully unrolled so wv stays in VGPRs
__device__ __forceinline__ float wsel(const float* wv, int d) {
    float r = 0.0f;
#pragma unroll
    for (int i = 0; i < 11; ++i) r = (d == i) ? wv[i] : r;
    return r;
}

__global__ __launch_bounds__(32)
void ssim_tiles(const float* __restrict__ x, const float* __restrict__ y,
                const float* __restrict__ win, float* __restrict__ partial) {
    // padded strides (33) keep LDS fragment gathers bank-conflict free
    __shared__ float xp[28][33];
    __shared__ float yp[28][33];
    __shared__ float sp[28][33];
    __shared__ float tb[16][33];

    const int lane = threadIdx.x;        // wave32: 0..31
    const int m    = lane & 15;          // row (A) or col (B/D) index
    const int half = lane >> 4;          // K-half select

    // 1D separable weights from 2D window: g[k] = k2d[k][5] / sqrt(k2d[5][5])
    float wv[11];
    {
        float g5 = sqrtf(win[5 * 11 + 5]);
#pragma unroll
        for (int k = 0; k < 11; ++k) wv[k] = win[k * 11 + 5] / g5;
    }

    // Banded weight fragments. A-side: A[m][k] = w[k-m] with
    // lanes0-15 -> K=4kc+{0,1}, lanes16-31 -> K=4kc+{2,3}.
    // B-side: B[k][n] = w[k-n] with the mirrored layout — identical values.
    v2f wfrag[KCH];
#pragma unroll
    for (int kc = 0; kc < KCH; ++kc) {
        int koff = 4 * kc + 2 * half;
        wfrag[kc][0] = wsel(wv, koff     - m);
        wfrag[kc][1] = wsel(wv, koff + 1 - m);
    }

    const float C1 = 1e-4f;   // 0.01^2
    const float C2 = 9e-4f;   // 0.03^2
    float acc_sum = 0.0f;

    for (int tloop = 0; tloop < TPB; ++tloop) {
        int tid   = blockIdx.x * TPB + tloop;
        int plane = tid >> 10;             // 32x32 = 1024 tiles per plane
        int rem   = tid & 1023;
        int ty    = rem >> 5, tx = rem & 31;
        int r0    = ty * 16,  c0 = tx * 16;
        const float* xb = x + (size_t)plane * IMG_H * IMG_W;
        const float* yb = y + (size_t)plane * IMG_H * IMG_W;

        // stage 28x32 halo patch (rows r0-5..r0+22, cols c0-5..c0+26), zero pad
        for (int i = lane; i < 28 * 32; i += 32) {
            int pr = i >> 5, pc = i & 31;
            int gr = r0 - 5 + pr, gc = c0 - 5 + pc;
            bool inb = (gr >= 0) & (gr < IMG_H) & (gc >= 0) & (gc < IMG_W);
            size_t idx = (size_t)gr * IMG_W + gc;
            xp[pr][pc] = inb ? xb[idx] : 0.0f;
            yp[pr][pc] = inb ? yb[idx] : 0.0f;
        }
        __syncthreads();

        v8f conv[5];
#pragma unroll
        for (int s = 0; s < 5; ++s) {
            // signal source: x and y read straight from the staged patches;
            // products are materialized into sp (s is compile-time constant)
            const float (*src)[33] = (s == 0) ? xp : (s == 1) ? yp : sp;
            if (s >= 2) {
                for (int i = lane; i < 28 * 32; i += 32) {
                    int pr = i >> 5, pc = i & 31;
                    float xv = xp[pr][pc], yv = yp[pr][pc];
                    float v = (s == 2) ? xv * xv
                            : (s == 3) ? yv * yv
                                       : xv * yv;
                    sp[pr][pc] = v;
                }
                __syncthreads();
            }

            // vertical pass: tmp(16x32) = Wband(16x28) x patch(28x32)
#pragma unroll
            for (int t = 0; t < 2; ++t) {
                v8f acc = {};
#pragma unroll
                for (int kc = 0; kc < KCH; ++kc) {
                    int krow = 4 * kc + 2 * half;
                    v2f b;
                    b[0] = src[krow    ][t * 16 + m];
                    b[1] = src[krow + 1][t * 16 + m];
                    acc = __builtin_amdgcn_wmma_f32_16x16x4_f32(
                        false, wfrag[kc], false, b, (short)0, acc, false, false);
                }
                // D layout: VGPR v, row = v + 8*half, col = m
#pragma unroll
                for (int v = 0; v < 8; ++v)
                    tb[v + 8 * half][t * 16 + m] = acc[v];
            }
            __syncthreads();

            // horizontal pass: out(16x16) = tmp(16x28) x Wband(28x16)
            {
                v8f acc = {};
#pragma unroll
                for (int kc = 0; kc < KCH; ++kc) {
                    int koff = 4 * kc + 2 * half;
                    v2f a;
                    a[0] = tb[m][koff];
                    a[1] = tb[m][koff + 1];
                    acc = __builtin_amdgcn_wmma_f32_16x16x4_f32(
                        false, a, false, wfrag[kc], (short)0, acc, false, false);
                }
                conv[s] = acc;
            }
            __syncthreads();
        }

        // elementwise SSIM map + accumulate (layout-independent per lane)
#pragma unroll
        for (int e = 0; e < 8; ++e) {
            float mx  = conv[0][e], my = conv[1][e];
            float mx2 = mx * mx, my2 = my * my, mxy = mx * my;
            float vx  = conv[2][e] - mx2;
            float vy  = conv[3][e] - my2;
            float vxy = conv[4][e] - mxy;
            float num = (2.0f * mxy + C1) * (2.0f * vxy + C2);
            float den = (mx2 + my2 + C1) * (vx + vy + C2);
            acc_sum += num / den;
        }
    }

    // wave32 reduction, deterministic, one partial per block
#pragma unroll
    for (int off = 16; off > 0; off >>= 1)
        acc_sum += __shfl_down(acc_sum, off, 32);
    if (lane == 0) partial[blockIdx.x] = acc_sum;
}

__global__ __launch_bounds__(256)
void ssim_reduce(const float* __restrict__ partial, int n, float* __restrict__ out) {
    __shared__ float sm[256];
    float s = 0.0f;
    for (int i = threadIdx.x; i < n; i += 256) s += partial[i];  // fixed order
    sm[threadIdx.x] = s;
    __syncthreads();
    for (int stride = 128; stride > 0; stride >>= 1) {
        if (threadIdx.x < stride) sm[threadIdx.x] += sm[threadIdx.x + stride];
        __syncthreads();
    }
    if (threadIdx.x == 0)
        out[0] = sm[0] * (1.0f / (float)((size_t)NC * IMG_H * IMG_W));
}

extern "C" void kernel_launch(void* const* d_in, const int* in_sizes, int n_in,
                              void* d_out, int out_size, void* d_ws, size_t ws_size,
                              hipStream_t stream) {
    const float* x   = (const float*)d_in[0];
    const float* y   = (const float*)d_in[1];
    const float* win = (const float*)d_in[2];   // (3,1,11,11); channel 0 at offset 0
    float* out = (float*)d_out;
    float* partial = (float*)d_ws;

    const int tiles  = NC * (IMG_H / 16) * (IMG_W / 16);  // 98304
    const int blocks = tiles / TPB;                        // 12288 (49 KB of d_ws)

    ssim_tiles<<<blocks, 32, 0, stream>>>(x, y, win, partial);
    ssim_reduce<<<1, 256, 0, stream>>>(partial, blocks, out);
}